// GraphSAGE_68264210202656
// MI455X (gfx1250) — compile-verified
//
#include <hip/hip_runtime.h>
#include <hip/hip_bf16.h>

// ---------------------------------------------------------------------------
// GraphSAGE forward for MI455X (gfx1250, wave32).
//   layer1: h   = relu(mean_agg(x) @ W1l + x @ W1r + b1)
//   layer2: emb =      mean_agg(h) @ W2l + h @ W2r + b2
//   out    = emb @ Wm + bm ;  emb_n = emb / max(||emb||, 1e-12)
// d_out = [ out (N*64) | emb_n (N*128) ]
// GEMMs use V_WMMA_F32_16X16X4_F32 (full fp32 precision; FLOPs are trivial
// next to the 3+ GB of edge scatter/gather traffic, which bounds the kernel).
// ---------------------------------------------------------------------------

typedef __attribute__((ext_vector_type(2))) float v2f;
typedef __attribute__((ext_vector_type(8))) float v8f;

#define N_NODES 100000
#define N_EDGES 1600000
#define CH      128
#define OUT_CH  64

__device__ __forceinline__ v8f wmma_f32_k4(v2f a, v2f b, v8f c) {
    // 8 args: (neg_a, A, neg_b, B, c_mod, C, reuse_a, reuse_b)
    return __builtin_amdgcn_wmma_f32_16x16x4_f32(
        false, a, false, b, (short)0, c, false, false);
}

// ---------------------------------------------------------------- zero fill
__global__ void zero_kernel(float* __restrict__ p, size_t n) {
    size_t i = (size_t)blockIdx.x * blockDim.x + threadIdx.x;
    if (i < n) p[i] = 0.0f;
}

// ------------------------------------------------------- edge scatter (mean)
// One wave per edge: coalesced float4 gather of feat[src], 4 f32 atomics per
// lane into sum[dst]. Sum buffer (51 MB) lives in the 192 MB L2, so atomics
// mostly resolve on-chip. cnt != nullptr only for layer 1 (degree reused).
__global__ void scatter_add_kernel(const float* __restrict__ feat,
                                   const int*   __restrict__ src,
                                   const int*   __restrict__ dst,
                                   float* __restrict__ sum,
                                   float* __restrict__ cnt,
                                   int nedges) {
    const int lane = threadIdx.x & 31;
    const int e = blockIdx.x * (blockDim.x >> 5) + (threadIdx.x >> 5);
    if (e >= nedges) return;
    const int s = src[e];
    const int d = dst[e];
    const float4 v = *(const float4*)(feat + (size_t)s * CH + lane * 4);
    float* p = sum + (size_t)d * CH + lane * 4;
    atomicAdd(p + 0, v.x);
    atomicAdd(p + 1, v.y);
    atomicAdd(p + 2, v.z);
    atomicAdd(p + 3, v.w);
    if (cnt != nullptr && lane == 0) atomicAdd(cnt + d, 1.0f);
}

// --------------------------------------------------------- sum -> mean rows
__global__ void mean_div_kernel(float* __restrict__ sum,
                                const float* __restrict__ cnt,
                                int nrows) {
    size_t idx = (size_t)blockIdx.x * blockDim.x + threadIdx.x; // 1 float4 each
    size_t total = (size_t)nrows * (CH / 4);
    if (idx >= total) return;
    const int row = (int)(idx >> 5);               // CH/4 == 32 float4 per row
    const float inv = 1.0f / fmaxf(cnt[row], 1.0f);
    float4* p = (float4*)sum + idx;
    float4 v = *p;
    v.x *= inv; v.y *= inv; v.z *= inv; v.w *= inv;
    *p = v;
}

// ------------------------------------------------------ fused SAGE-GEMM tile
// out[r][c] = act( A0[r,:]@W0[:,c] (+ A1[r,:]@W1[:,c]) + bias[c] )
// One wave per 16x16 tile; blockDim.x = 32 * (ncols/16); grid.x = nrows/16.
// K fixed at 128. nrows must be a multiple of 16 (100000 is) so EXEC==all 1s.
template <bool TWO, bool RELU>
__global__ void sage_gemm_kernel(const float* __restrict__ A0,
                                 const float* __restrict__ A1,
                                 const float* __restrict__ W0,
                                 const float* __restrict__ W1,
                                 const float* __restrict__ bias,
                                 float* __restrict__ out,
                                 int ncols) {
    const int lane = threadIdx.x & 31;
    const int wave = threadIdx.x >> 5;
    const int row0 = blockIdx.x * 16;
    const int col0 = wave * 16;
    // A-matrix 16x4 f32 layout: lane holds A[row0 + lane%16][k + koff, k+koff+1]
    const int r    = row0 + (lane & 15);
    const int koff = (lane >> 4) * 2;              // 0 (lanes 0-15) / 2 (16-31)
    // B-matrix 4x16 f32 layout (mirrors A): lane holds W[k+koff..+1][col]
    const int c    = col0 + (lane & 15);

    v8f acc = {};
    for (int k = 0; k < CH; k += 4) {
        const int kk = k + koff;
        v2f a0 = *(const v2f*)(A0 + (size_t)r * CH + kk);
        v2f b0;
        b0.x = W0[(size_t)kk * ncols + c];
        b0.y = W0[(size_t)(kk + 1) * ncols + c];
        acc = wmma_f32_k4(a0, b0, acc);
        if (TWO) {
            v2f a1 = *(const v2f*)(A1 + (size_t)r * CH + kk);
            v2f b1;
            b1.x = W1[(size_t)kk * ncols + c];
            b1.y = W1[(size_t)(kk + 1) * ncols + c];
            acc = wmma_f32_k4(a1, b1, acc);
        }
    }

    const float bv = bias[c];
    // C/D layout: VGPR v -> row = row0 + v + (lane>=16 ? 8 : 0), col = c
    const int rbase = row0 + ((lane >> 4) << 3);
#pragma unroll
    for (int v = 0; v < 8; ++v) {
        float val = acc[v] + bv;
        if (RELU) val = fmaxf(val, 0.0f);
        out[(size_t)(rbase + v) * ncols + c] = val;
    }
}

// ------------------------------------------------------ in-place L2 normalize
__global__ void l2norm_kernel(float* __restrict__ emb, int nrows) {
    const int lane = threadIdx.x & 31;
    const int row = blockIdx.x * (blockDim.x >> 5) + (threadIdx.x >> 5);
    if (row >= nrows) return;
    float4* p = (float4*)(emb + (size_t)row * CH) + lane;
    float4 v = *p;
    float ss = v.x * v.x + v.y * v.y + v.z * v.z + v.w * v.w;
#pragma unroll
    for (int o = 16; o > 0; o >>= 1) ss += __shfl_xor(ss, o, 32);
    const float inv = 1.0f / fmaxf(sqrtf(ss), 1e-12f);
    v.x *= inv; v.y *= inv; v.z *= inv; v.w *= inv;
    *p = v;
}

// ---------------------------------------------------------------------------
extern "C" void kernel_launch(void* const* d_in, const int* in_sizes, int n_in,
                              void* d_out, int out_size, void* d_ws, size_t ws_size,
                              hipStream_t stream) {
    const float* x   = (const float*)d_in[0];
    const int*   ei  = (const int*)  d_in[1];
    const float* W1l = (const float*)d_in[2];
    const float* b1  = (const float*)d_in[3];
    const float* W1r = (const float*)d_in[4];
    const float* W2l = (const float*)d_in[5];
    const float* b2  = (const float*)d_in[6];
    const float* W2r = (const float*)d_in[7];
    const float* Wm  = (const float*)d_in[8];
    const float* bm  = (const float*)d_in[9];

    const int N = N_NODES;
    const int E = in_sizes[1] / 2;
    const int* src  = ei;
    const int* dstI = ei + E;

    float* ws  = (float*)d_ws;
    float* sum = ws;                          // N*128
    float* cnt = sum + (size_t)N * CH;        // N (contiguous after sum)
    float* h   = cnt + N;                     // N*128

    float* outp = (float*)d_out;                        // N*64
    float* emb  = (float*)d_out + (size_t)N * OUT_CH;   // N*128

    const int rowTiles = N / 16;              // 6250

    // ---- layer 1 aggregation ----
    {   // zero sum + cnt together (contiguous N*(128+1) floats)
        size_t nz = (size_t)N * (CH + 1);
        zero_kernel<<<dim3((unsigned)((nz + 255) / 256)), 256, 0, stream>>>(sum, nz);
    }
    {   // scatter x[src] into sum[dst], count degrees
        unsigned blocks = (unsigned)((E + 7) / 8);      // 8 edges/block @ 256 thr
        scatter_add_kernel<<<blocks, 256, 0, stream>>>(x, src, dstI, sum, cnt, E);
    }
    {   size_t nq = (size_t)N * (CH / 4);
        mean_div_kernel<<<dim3((unsigned)((nq + 255) / 256)), 256, 0, stream>>>(sum, cnt, N);
    }
    // h = relu(mean @ W1l + x @ W1r + b1)   [8 waves/block, 16 rows x 128 cols]
    sage_gemm_kernel<true, true><<<rowTiles, 256, 0, stream>>>(
        sum, x, W1l, W1r, b1, h, CH);

    // ---- layer 2 aggregation (degrees reused) ----
    {   size_t nz = (size_t)N * CH;
        zero_kernel<<<dim3((unsigned)((nz + 255) / 256)), 256, 0, stream>>>(sum, nz);
    }
    {   unsigned blocks = (unsigned)((E + 7) / 8);
        scatter_add_kernel<<<blocks, 256, 0, stream>>>(h, src, dstI, sum, nullptr, E);
    }
    {   size_t nq = (size_t)N * (CH / 4);
        mean_div_kernel<<<dim3((unsigned)((nq + 255) / 256)), 256, 0, stream>>>(sum, cnt, N);
    }
    // emb = mean2 @ W2l + h @ W2r + b2  (written straight into d_out slot)
    sage_gemm_kernel<true, false><<<rowTiles, 256, 0, stream>>>(
        sum, h, W2l, W2r, b2, emb, CH);

    // ---- head: out = emb @ Wm + bm   [4 waves/block, 16 rows x 64 cols] ----
    sage_gemm_kernel<false, false><<<rowTiles, 128, 0, stream>>>(
        emb, nullptr, Wm, nullptr, bm, outp, OUT_CH);

    // ---- emb_n = emb / max(||emb||, 1e-12), in place (after head reads it) ----
    l2norm_kernel<<<(N + 7) / 8, 256, 0, stream>>>(emb, N);
}